// Histogram2D_48558900249123
// MI455X (gfx1250) — compile-verified
//
#include <hip/hip_runtime.h>

#define BINS   100
#define NBINS2 (BINS * BINS)
#define KOFF   2
#define DELTA  0.01f
#define NBATCH 16
#define WGX    32   // workgroups per batch for the histogram pass

typedef __attribute__((ext_vector_type(2))) float v2f;
typedef __attribute__((ext_vector_type(8))) float v8f;

// ---------------------------------------------------------------------------
// Kernel 1: zero the per-batch accumulation buffer in d_ws (harness poisons it)
// ---------------------------------------------------------------------------
__global__ void Histo2D_zero_kernel(float* __restrict__ g, int n) {
    int i = blockIdx.x * blockDim.x + threadIdx.x;
    if (i < n) g[i] = 0.0f;
}

// ---------------------------------------------------------------------------
// Kernel 2: LDS-privatized 2D soft histogram.
//   grid = (WGX, NBATCH), block = 256 (8 wave32s).
//   40 KB LDS histogram per block (320 KB/WGP -> up to 8 blocks resident).
// ---------------------------------------------------------------------------
__global__ __launch_bounds__(256) void Histo2D_hist_kernel(
    const float* __restrict__ x, float* __restrict__ ghist, int N) {

    __shared__ float lhist[NBINS2];                       // 40,000 bytes
    const int tid = threadIdx.x;

    for (int i = tid; i < NBINS2; i += blockDim.x) lhist[i] = 0.0f;
    __syncthreads();

    const int batch = blockIdx.y;
    const float2* __restrict__ xb =
        (const float2*)(x + (size_t)batch * (size_t)N * 2);
    const int stride = gridDim.x * blockDim.x;

    for (int n = blockIdx.x * blockDim.x + tid; n < N; n += stride) {
        // speculative prefetch of next tile (global_prefetch_b8; OOB is dropped)
        __builtin_prefetch(&xb[n + stride], 0, 0);

        float2 p = xb[n];

        float d0[5], d1[5];
        int   j0[5], j1[5];

        int i0 = (int)floorf(p.x * (float)BINS);
        i0 = min(max(i0, 0), BINS - 1);
        int i1 = (int)floorf(p.y * (float)BINS);
        i1 = min(max(i1, 0), BINS - 1);

        #pragma unroll
        for (int o = 0; o < 5; ++o) {
            int  nb = i0 + o - KOFF;
            bool ok = (nb >= 0) && (nb < BINS);
            int  nc = min(max(nb, 0), BINS - 1);
            j0[o] = nc;
            d0[o] = ok ? fabsf(p.x - ((float)nc + 0.5f) * DELTA) : 1.0e9f;

            nb = i1 + o - KOFF;
            ok = (nb >= 0) && (nb < BINS);
            nc = min(max(nb, 0), BINS - 1);
            j1[o] = nc;
            d1[o] = ok ? fabsf(p.y - ((float)nc + 0.5f) * DELTA) : 1.0e9f;
        }

        #pragma unroll
        for (int a = 0; a < 5; ++a) {
            #pragma unroll
            for (int b = 0; b < 5; ++b) {
                float w = DELTA - 0.5f * (d0[a] + d1[b]);
                if (w > 0.0f) {                            // relu + skip zero taps
                    atomicAdd(&lhist[j0[a] * BINS + j1[b]], w);   // ds_add_f32
                }
            }
        }
    }
    __syncthreads();

    // Flush private histogram into the per-batch global accumulator.
    float* __restrict__ gh = ghist + batch * NBINS2;
    for (int i = tid; i < NBINS2; i += blockDim.x) {
        float v = lhist[i];
        if (v != 0.0f) atomicAdd(&gh[i], v);               // global_atomic_add_f32
    }
}

// ---------------------------------------------------------------------------
// Kernel 3: per-batch normalization. grid = 16, block = 1024 (32 wave32s).
//   Per-wave shfl_xor reduction, then the 32 wave partials are reduced with a
//   single exact f32 V_WMMA_F32_16X16X4_F32 (A = partials, B = ones):
//   D[m,n] = p[m] + p[m+16]; 8 adds + one xor-16 shuffle -> block total in
//   every lane of every wave (full EXEC at the WMMA as required).
// ---------------------------------------------------------------------------
__global__ __launch_bounds__(1024) void Histo2D_norm_kernel(
    const float* __restrict__ ghist, float* __restrict__ out) {

    const int batch = blockIdx.x;
    const float* __restrict__ gh = ghist + batch * NBINS2;
    const int tid  = threadIdx.x;
    const int lane = tid & 31;
    const int wid  = tid >> 5;

    float partial = 0.0f;
    for (int i = tid; i < NBINS2; i += blockDim.x) partial += gh[i];

    #pragma unroll
    for (int off = 16; off >= 1; off >>= 1)
        partial += __shfl_xor(partial, off, 32);

    __shared__ float wsum[32];
    if (lane == 0) wsum[wid] = partial;
    __syncthreads();

    // All waves redundantly reduce the 32 wave-partials (EXEC all ones).
    float v = wsum[lane];
    v2f A;  A.x = v;    A.y = 0.0f;     // A[m,0]=p[m], A[m,2]=p[m+16], rest 0
    v2f Bm; Bm.x = 1.0f; Bm.y = 1.0f;   // B = ones(4x16)
    v8f C = {};
    v8f D = __builtin_amdgcn_wmma_f32_16x16x4_f32(
        /*neg_a=*/false, A, /*neg_b=*/false, Bm,
        /*c_mod=*/(short)0, C, /*reuse_a=*/false, /*reuse_b=*/false);

    float s = D[0] + D[1] + D[2] + D[3] + D[4] + D[5] + D[6] + D[7];
    s += __shfl_xor(s, 16, 32);         // total of all 32 partials, every lane

    const float inv = 1.0f / (s + 1e-5f);
    float* __restrict__ ob = out + batch * NBINS2;
    for (int i = tid; i < NBINS2; i += blockDim.x) ob[i] = gh[i] * inv;
}

// ---------------------------------------------------------------------------
extern "C" void kernel_launch(void* const* d_in, const int* in_sizes, int n_in,
                              void* d_out, int out_size, void* d_ws, size_t ws_size,
                              hipStream_t stream) {
    const float* x   = (const float*)d_in[0];
    float*       out = (float*)d_out;
    float*       gh  = (float*)d_ws;            // NBATCH * 10000 f32 = 640 KB

    const int N = in_sizes[0] / (NBATCH * 2);   // 131072 samples per batch

    const int ztot = NBATCH * NBINS2;
    Histo2D_zero_kernel<<<(ztot + 255) / 256, 256, 0, stream>>>(gh, ztot);

    dim3 grid(WGX, NBATCH);
    Histo2D_hist_kernel<<<grid, 256, 0, stream>>>(x, gh, N);

    Histo2D_norm_kernel<<<NBATCH, 1024, 0, stream>>>(gh, out);
}